// DownConvPoint_58969900974257
// MI455X (gfx1250) — compile-verified
//
#include <hip/hip_runtime.h>
#include <hip/hip_bf16.h>

typedef __attribute__((ext_vector_type(2))) float v2f;
typedef __attribute__((ext_vector_type(8))) float v8f;

#define NBR_K 6
#define COUT 128
#define USE_ASYNC_GATHER 1

// One-element global->LDS copy. Async path: CDNA5 GLOBAL_LOAD_ASYNC_TO_LDS_B32
// (per-lane addresses, tracked by ASYNCcnt, no VGPR round-trip).
__device__ __forceinline__ void g2l_b32(float* lds, const float* g) {
#if USE_ASYNC_GATHER
  asm volatile("global_load_async_to_lds_b32 %0, %1, off"
               :
               : "v"((unsigned)(uintptr_t)lds),          // LDS byte offset (low 32 bits)
                 "v"((unsigned long long)(uintptr_t)g)   // 64-bit global address
               : "memory");
#else
  *lds = *g;
#endif
}

__device__ __forceinline__ void gather_wait() {
#if USE_ASYNC_GATHER
  asm volatile("s_wait_asynccnt 0" ::: "memory");
#endif
}

// ---------------------------------------------------------------------------
// Fused mesh-conv GEMM with f32 WMMA.
//   y[b, o, v] = sum_{c,k} x[b,c,v,k] * w[o, c*7+k] + bias[o]
// One workgroup = one batch b, 64 vertices, all 128 output channels.
// 8 waves; wave w owns output rows [16w, 16w+16) x 64 vertices (4 N-tiles).
// K-dim (CIN*7) processed in chunks of 112 (16 channels x 7 taps) via LDS.
// LDS layout interleaves K-row pairs so each lane's B-fragment (K even, K odd)
// is one contiguous 8-byte ds_load_b64:
//   element(row r, col c) -> Xl[(r>>1)*132 + 2*c + (r&1)]
// Epilogue: add bias, store y, and accumulate per-(b,o) sum / sumsq with a
// 16-lane butterfly + global atomicAdd (for the InstanceNorm that follows).
// ---------------------------------------------------------------------------
template <int CIN>
__global__ __launch_bounds__(256) void meshconv_wmma(
    const float* __restrict__ f,     // [B, CIN, V]
    const int* __restrict__ nbr,     // [B, V, 6]
    const float* __restrict__ w,     // [128, CIN*7]
    const float* __restrict__ bias,  // [128]
    float* __restrict__ y,           // [B, 128, V]
    float* __restrict__ osum,        // [B*128]
    float* __restrict__ osumsq,      // [B*128]
    int V) {
  constexpr int CK = CIN * 7;
  constexpr int NCHUNK = CIN / 16;     // 112-wide K chunks
  constexpr int PAIRP = 132;           // floats per K-row-pair (64 cols*2 + pad)
  __shared__ float Xl[56 * PAIRP];     // 29.6 KB: 112 K-rows x 64 cols, pair-interleaved

  const int b = blockIdx.y;
  const int v0 = blockIdx.x * 64;
  int nvert = V - v0;
  if (nvert > 64) nvert = 64;
  const int nvt = nvert >> 4;          // valid 16-vertex N-tiles (1..4)

  const int tid = threadIdx.x;
  const int wave = tid >> 5;
  const int lane = tid & 31;
  const int half = lane >> 4;          // 0: K0/K1 lanes, 1: K2/K3 lanes
  const int l15 = lane & 15;

  // --- per-thread gather column descriptors (col = vv*7 + k, 448 columns) ---
  const float* fb = f + (size_t)b * CIN * V;
  int src0 = -1, src1 = -1;
  int vv0 = 0, kk0 = 0, vv1 = 0, kk1 = 0;
  {
    int col = tid;                     // 0..255
    vv0 = col / 7; kk0 = col - vv0 * 7;
    if (vv0 < nvert) {
      int v = v0 + vv0;
      src0 = (kk0 == 0) ? v : nbr[((size_t)b * V + v) * NBR_K + (kk0 - 1)];
    }
    col = tid + 256;                   // 256..511; columns exist up to 447
    if (col < 448) {
      vv1 = col / 7; kk1 = col - vv1 * 7;
      if (vv1 < nvert) {
        int v = v0 + vv1;
        src1 = (kk1 == 0) ? v : nbr[((size_t)b * V + v) * NBR_K + (kk1 - 1)];
      }
    }
  }

  v8f acc0 = {}, acc1 = {}, acc2 = {}, acc3 = {};
  // A-fragment source row for this lane (W row m, two K values per lane)
  const float* wrow = w + (size_t)(wave * 16 + l15) * CK + 2 * half;

  for (int ch = 0; ch < NCHUNK; ++ch) {
    __syncthreads();                   // previous chunk fully consumed
    const int cbase = ch * 16;
    if (src0 >= 0) {
      const float* g = fb + (size_t)cbase * V + src0;
#pragma unroll
      for (int cc = 0; cc < 16; ++cc) {
        const int r = cc * 7 + kk0;
        g2l_b32(&Xl[(r >> 1) * PAIRP + 2 * vv0 + (r & 1)], g + (size_t)cc * V);
      }
    }
    if (src1 >= 0) {
      const float* g = fb + (size_t)cbase * V + src1;
#pragma unroll
      for (int cc = 0; cc < 16; ++cc) {
        const int r = cc * 7 + kk1;
        g2l_b32(&Xl[(r >> 1) * PAIRP + 2 * vv1 + (r & 1)], g + (size_t)cc * V);
      }
    }
    gather_wait();                     // drain this wave's ASYNCcnt
    __syncthreads();                   // all waves' gathers visible

    const float* wr = wrow + ch * 112;
#pragma unroll 4
    for (int kb = 0; kb < 112; kb += 4) {
      // A: 16x4 of W (lanes 0-15: K=kb,kb+1; lanes 16-31: K=kb+2,kb+3)
      v2f a = *(const v2f*)(wr + kb);  // 8B aligned: even float offset
      // B: one ds_load_b64 per N-tile (K even, K odd contiguous)
      const float* xp = &Xl[((kb + 2 * half) >> 1) * PAIRP + 2 * l15];
      v2f bm;
      bm = *(const v2f*)(xp + 0);
      acc0 = __builtin_amdgcn_wmma_f32_16x16x4_f32(false, a, false, bm,
                                                   (short)0, acc0, false, false);
      bm = *(const v2f*)(xp + 32);
      acc1 = __builtin_amdgcn_wmma_f32_16x16x4_f32(false, a, false, bm,
                                                   (short)0, acc1, false, false);
      bm = *(const v2f*)(xp + 64);
      acc2 = __builtin_amdgcn_wmma_f32_16x16x4_f32(false, a, false, bm,
                                                   (short)0, acc2, false, false);
      bm = *(const v2f*)(xp + 96);
      acc3 = __builtin_amdgcn_wmma_f32_16x16x4_f32(false, a, false, bm,
                                                   (short)0, acc3, false, false);
    }
  }

  // --- epilogue: bias, store, per-row sum/sumsq for instance norm ----------
#pragma unroll
  for (int j = 0; j < 8; ++j) {
    const int mg = wave * 16 + 8 * half + j;       // D layout: M = j + 8*half
    const float bb = bias[mg];
    float* yr = y + ((size_t)b * COUT + mg) * V + v0 + l15;
    float s = 0.f, q = 0.f, t;
    t = acc0[j] + bb;               { yr[0]  = t; s += t; q += t * t; }
    t = acc1[j] + bb; if (nvt > 1)  { yr[16] = t; s += t; q += t * t; }
    t = acc2[j] + bb; if (nvt > 2)  { yr[32] = t; s += t; q += t * t; }
    t = acc3[j] + bb; if (nvt > 3)  { yr[48] = t; s += t; q += t * t; }
#pragma unroll
    for (int mask = 8; mask >= 1; mask >>= 1) {
      s += __shfl_xor(s, mask, 32);
      q += __shfl_xor(q, mask, 32);
    }
    if (l15 == 0) {
      atomicAdd(&osum[b * COUT + mg], s);
      atomicAdd(&osumsq[b * COUT + mg], q);
    }
  }
}

// In-place InstanceNorm + ReLU:  y = relu((y - mean) * rsqrt(var + eps))
__global__ __launch_bounds__(256) void norm_relu(
    float* __restrict__ y, const float* __restrict__ osum,
    const float* __restrict__ osumsq, int V, float invV) {
  const int row = blockIdx.y;                     // b*128 + o
  const int v = blockIdx.x * 256 + threadIdx.x;
  if (v >= V) return;
  const float m = osum[row] * invV;
  const float var = osumsq[row] * invV - m * m;
  const float r = rsqrtf(var + 1e-5f);
  const size_t i = (size_t)row * V + v;
  const float t = (y[i] - m) * r;
  y[i] = t > 0.f ? t : 0.f;
}

// In-place InstanceNorm + residual + ReLU: y2 = relu(norm(y2) + x1)
__global__ __launch_bounds__(256) void norm_add_relu(
    float* __restrict__ y2, const float* __restrict__ x1,
    const float* __restrict__ osum, const float* __restrict__ osumsq,
    int V, float invV) {
  const int row = blockIdx.y;
  const int v = blockIdx.x * 256 + threadIdx.x;
  if (v >= V) return;
  const float m = osum[row] * invV;
  const float var = osumsq[row] * invV - m * m;
  const float r = rsqrtf(var + 1e-5f);
  const size_t i = (size_t)row * V + v;
  const float t = (y2[i] - m) * r + x1[i];
  y2[i] = t > 0.f ? t : 0.f;
}

__global__ void zero_f32(float* __restrict__ p, int n) {
  const int i = blockIdx.x * 256 + threadIdx.x;
  if (i < n) p[i] = 0.f;
}

extern "C" void kernel_launch(void* const* d_in, const int* in_sizes, int n_in,
                              void* d_out, int out_size, void* d_ws, size_t ws_size,
                              hipStream_t stream) {
  const float* fe = (const float*)d_in[0];   // [4, 64, 50000]
  const int* nbrs = (const int*)d_in[1];     // [4, 50000, 6]
  const float* w1 = (const float*)d_in[2];   // [128, 64, 7]
  const float* b1 = (const float*)d_in[3];   // [128]
  const float* w2 = (const float*)d_in[4];   // [128, 128, 7]
  const float* b2 = (const float*)d_in[5];   // [128]
  float* out = (float*)d_out;                // [4, 128, 50000]

  const int B = 4, V = 50000;
  float* y1 = (float*)d_ws;                  // [4, 128, V] = 102.4 MB
  float* stats = y1 + (size_t)B * COUT * V;  // 4 x 512 floats
  float* sum1 = stats;
  float* sq1 = stats + 512;
  float* sum2 = stats + 1024;
  float* sq2 = stats + 1536;

  zero_f32<<<8, 256, 0, stream>>>(stats, 2048);

  dim3 gconv((V + 63) / 64, B);              // 782 x 4 tiles
  dim3 gnorm((V + 255) / 256, B * COUT);     // 196 x 512

  // conv1 -> y1 (+ norm stats)
  meshconv_wmma<64><<<gconv, 256, 0, stream>>>(fe, nbrs, w1, b1, y1, sum1, sq1, V);
  // instance norm + relu in place -> x1 lives in y1
  norm_relu<<<gnorm, 256, 0, stream>>>(y1, sum1, sq1, V, 1.0f / V);
  // conv2 on x1 -> d_out (+ norm stats)
  meshconv_wmma<128><<<gconv, 256, 0, stream>>>(y1, nbrs, w2, b2, out, sum2, sq2, V);
  // instance norm + residual + relu in place on d_out
  norm_add_relu<<<gnorm, 256, 0, stream>>>(out, y1, sum2, sq2, V, 1.0f / V);
}